// Splatter_70248485093630
// MI455X (gfx1250) — compile-verified
//
#include <hip/hip_runtime.h>
#include <math.h>

typedef float v2f __attribute__((ext_vector_type(2)));
typedef float v8f __attribute__((ext_vector_type(8)));

#define WIDTH   512
#define HEIGHT  512
#define NTX     32
#define NTILES  1024
#define KPT     128
#define CAP     6144   // LDS candidate capacity per tile (48KB)

__device__ __forceinline__ void quat_rot(float w, float x, float y, float z, float R[3][3]) {
  float inv = rsqrtf(w*w + x*x + y*y + z*z);
  w *= inv; x *= inv; y *= inv; z *= inv;
  R[0][0] = 1.f - 2.f*(y*y + z*z); R[0][1] = 2.f*(x*y - w*z);       R[0][2] = 2.f*(x*z + w*y);
  R[1][0] = 2.f*(x*y + w*z);       R[1][1] = 1.f - 2.f*(x*x + z*z); R[1][2] = 2.f*(y*z - w*x);
  R[2][0] = 2.f*(x*z - w*y);       R[2][1] = 2.f*(y*z + w*x);       R[2][2] = 1.f - 2.f*(x*x + y*y);
}

// ---------------------------------------------------------------------------
// Kernel 1: per-gaussian preprocess -> gsel (for selection), gA/gIC (conic)
// ---------------------------------------------------------------------------
__global__ __launch_bounds__(256)
void k_preprocess(const float* __restrict__ pos, const float* __restrict__ quat,
                  const float* __restrict__ scl, const float* __restrict__ w2cq,
                  const float* __restrict__ w2ct, int N,
                  float4* __restrict__ gsel, float4* __restrict__ gA,
                  float* __restrict__ gIC)
{
  int n = blockIdx.x * 256 + threadIdx.x;
  if (n >= N) return;

  float Rc[3][3];
  quat_rot(w2cq[0], w2cq[1], w2cq[2], w2cq[3], Rc);

  float px = pos[3*n+0], py = pos[3*n+1], pz = pos[3*n+2];
  float x = Rc[0][0]*px + Rc[0][1]*py + Rc[0][2]*pz + w2ct[0];
  float y = Rc[1][0]*px + Rc[1][1]*py + Rc[1][2]*pz + w2ct[1];
  float z = Rc[2][0]*px + Rc[2][1]*py + Rc[2][2]*pz + w2ct[2];

  const float fx = (float)WIDTH, fy = (float)HEIGHT;
  float zc = fmaxf(z, 1e-6f);
  bool valid = (z > 0.3f) && (fabsf(x/zc) < 0.6f) && (fabsf(y/zc) < 0.6f);

  float Rg[3][3];
  quat_rot(quat[4*n+0], quat[4*n+1], quat[4*n+2], quat[4*n+3], Rg);
  float s0 = 1.f / (1.f + __expf(-scl[3*n+0]));
  float s1 = 1.f / (1.f + __expf(-scl[3*n+1]));
  float s2 = 1.f / (1.f + __expf(-scl[3*n+2]));

  // M = Rg * diag-ish(s);  Sigma = M M^T
  float M0[3] = { Rg[0][0]*s0, Rg[0][1]*s1, Rg[0][2]*s2 };
  float M1[3] = { Rg[1][0]*s0, Rg[1][1]*s1, Rg[1][2]*s2 };
  float M2[3] = { Rg[2][0]*s0, Rg[2][1]*s1, Rg[2][2]*s2 };
  float S[3][3];
  S[0][0] = M0[0]*M0[0] + M0[1]*M0[1] + M0[2]*M0[2];
  S[0][1] = M0[0]*M1[0] + M0[1]*M1[1] + M0[2]*M1[2];
  S[0][2] = M0[0]*M2[0] + M0[1]*M2[1] + M0[2]*M2[2];
  S[1][1] = M1[0]*M1[0] + M1[1]*M1[1] + M1[2]*M1[2];
  S[1][2] = M1[0]*M2[0] + M1[1]*M2[1] + M1[2]*M2[2];
  S[2][2] = M2[0]*M2[0] + M2[1]*M2[1] + M2[2]*M2[2];
  S[1][0] = S[0][1]; S[2][0] = S[0][2]; S[2][1] = S[1][2];

  // Sigma_cam = Rc S Rc^T
  float Tm[3][3], SC[3][3];
  for (int i = 0; i < 3; i++)
    for (int k = 0; k < 3; k++)
      Tm[i][k] = Rc[i][0]*S[0][k] + Rc[i][1]*S[1][k] + Rc[i][2]*S[2][k];
  for (int i = 0; i < 3; i++)
    for (int l = 0; l < 3; l++)
      SC[i][l] = Tm[i][0]*Rc[l][0] + Tm[i][1]*Rc[l][1] + Tm[i][2]*Rc[l][2];

  float j00 = fx/zc, j02 = -fx*x/(zc*zc);
  float j11 = fy/zc, j12 = -fy*y/(zc*zc);
  // v = SC * j0^T  with j0 = (j00, 0, j02)
  float v0x = SC[0][0]*j00 + SC[0][2]*j02;
  float v0y = SC[1][0]*j00 + SC[1][2]*j02;
  float v0z = SC[2][0]*j00 + SC[2][2]*j02;
  float a = j00*v0x + j02*v0z + 0.3f;
  float b = j11*v0y + j12*v0z;
  float u1 = SC[1][1]*j11 + SC[1][2]*j12;
  float u2 = SC[2][1]*j11 + SC[2][2]*j12;
  float c = j11*u1 + j12*u2 + 0.3f;

  float det = a*c - b*b;
  float mid = 0.5f*(a + c);
  float radius = 3.f * sqrtf(mid + sqrtf(fmaxf(mid*mid - det, 0.01f)));
  float mx = fx*x/zc + 0.5f*(float)WIDTH;
  float my = fy*y/zc + 0.5f*(float)HEIGHT;
  float gdet = fmaxf(det, 1e-8f);
  float ia = c/gdet, ib = -b/gdet, ic = a/gdet;

  gsel[n] = make_float4(mx, my, radius + 11.3137085f, valid ? z : -1.0f);
  gA[n]   = make_float4(mx, my, ia, ib);
  gIC[n]  = ic;
}

// ---------------------------------------------------------------------------
// Kernel 2: per-tile coverage compaction + deterministic sorted top-128
// key = (depth_bits << 32) | idx  (positive fp32 bits are order-preserving;
// idx tie-break matches top_k's lowest-index rule)
// ---------------------------------------------------------------------------
__global__ __launch_bounds__(256)
void k_select(const float4* __restrict__ gsel, int N,
              int* __restrict__ sel_idx, int* __restrict__ sel_cnt)
{
  __shared__ unsigned long long cand[CAP];
  __shared__ unsigned long long s_gmin;
  __shared__ int s_cnt;

  int t = blockIdx.x, tid = threadIdx.x;
  float tcx = ((float)(t & (NTX-1)) + 0.5f) * 16.f;
  float tcy = ((float)(t >> 5)      + 0.5f) * 16.f;

  if (tid == 0) s_cnt = 0;
  __syncthreads();

  for (int g = tid; g < N; g += 256) {
    if (g + 1024 < N) __builtin_prefetch(gsel + g + 1024, 0, 0);
    float4 v = gsel[g];
    if (v.w > 0.f) {
      float dx = tcx - v.x, dy = tcy - v.y;
      if (dx*dx + dy*dy <= v.z*v.z) {
        unsigned long long pk =
            ((unsigned long long)__float_as_uint(v.w) << 32) | (unsigned)g;
        int p = atomicAdd(&s_cnt, 1);
        if (p < CAP) cand[p] = pk;
      }
    }
  }
  __syncthreads();

  int m = min(s_cnt, CAP);
  unsigned long long last = 0ull;
  int kcnt = KPT;
  for (int k = 0; k < KPT; k++) {
    if (tid == 0) s_gmin = ~0ull;
    __syncthreads();
    unsigned long long loc = ~0ull;
    for (int i = tid; i < m; i += 256) {
      unsigned long long ck = cand[i];
      if (ck > last && ck < loc) loc = ck;
    }
    if (loc != ~0ull) atomicMin(&s_gmin, loc);
    __syncthreads();
    unsigned long long win = s_gmin;
    __syncthreads();
    if (win == ~0ull) { kcnt = k; break; }
    if (tid == 0) sel_idx[t*KPT + k] = (int)(win & 0xffffffffull);
    last = win;
  }
  if (tid == 0) sel_cnt[t] = kcnt;
}

// ---------------------------------------------------------------------------
// Kernel 3: per-tile rasterization. Sequential transparency per pixel, then
// color = weight^T @ rgb via V_WMMA_F32_16X16X4_F32 (full fp32 matrix core).
// 8 waves/block, each wave owns 2 pixel rows (16 pixels each).
// A lane l: row M = l&15 (pixel column), lanes 0-15 hold K={k,k+1},
// lanes 16-31 hold K={k+2,k+3}. B mirrors the K striping; N = channel.
// Channel selection is a once-per-lane LDS base-pointer pick (no branches
// inside the WMMA loop).
// ---------------------------------------------------------------------------
__global__ __launch_bounds__(256)
void k_raster(const float4* __restrict__ gA, const float* __restrict__ gIC,
              const float* __restrict__ rgb, const float* __restrict__ opa,
              const int* __restrict__ sel_idx, const int* __restrict__ sel_cnt,
              float* __restrict__ out)
{
  __shared__ float smx[KPT], smy[KPT], sia[KPT], sib[KPT], sic[KPT];
  __shared__ float sr[KPT], sg[KPT], sb[KPT], sop[KPT];
  __shared__ float szero[KPT];

  int t = blockIdx.x, tid = threadIdx.x;
  int cnt = sel_cnt[t];
  if (tid < KPT) {
    szero[tid] = 0.f;
    if (tid < cnt) {
      int i = sel_idx[t*KPT + tid];
      float4 A = gA[i];
      smx[tid] = A.x; smy[tid] = A.y; sia[tid] = A.z; sib[tid] = A.w;
      sic[tid] = gIC[i];
      sr[tid] = rgb[3*i+0]; sg[tid] = rgb[3*i+1]; sb[tid] = rgb[3*i+2];
      sop[tid] = opa[i];
    } else {
      smx[tid] = 0.f; smy[tid] = 0.f; sia[tid] = 0.f; sib[tid] = 0.f;
      sic[tid] = 0.f; sr[tid] = 0.f;  sg[tid] = 0.f;  sb[tid] = 0.f;
      sop[tid] = 0.f;   // alpha == 0 for unselected slots (matches sel mask)
    }
  }
  __syncthreads();

  int lane = tid & 31;
  int wave = tid >> 5;
  int half = lane >> 4;        // 0: K pair {k,k+1}; 1: K pair {k+2,k+3}
  int nn   = lane & 15;        // A row (pixel column) and B/D column (channel)

  // Per-lane B source: channel array (or zeros), chosen once.
  const float* bsrc = (nn == 0) ? sr : (nn == 1) ? sg : (nn == 2) ? sb : szero;

  float tx0 = (float)((t & (NTX-1)) * 16);
  float ty0 = (float)((t >> 5)      * 16);

  for (int rb = 0; rb < 2; rb++) {
    int pb = wave + rb * 8;                 // pixel row within tile
    float pixx = tx0 + (float)nn + 0.5f;
    float pixy = ty0 + (float)pb + 0.5f;

    v8f acc = {0.f, 0.f, 0.f, 0.f, 0.f, 0.f, 0.f, 0.f};
    float trans = 1.f;

    for (int k = 0; k < KPT; k += 4) {
      float w0, w1, w2, w3;
      {
        float dx = pixx - smx[k+0], dy = pixy - smy[k+0];
        float pw = -0.5f*(sia[k+0]*dx*dx + 2.f*sib[k+0]*dx*dy + sic[k+0]*dy*dy);
        float al = fminf(0.99f, sop[k+0] * __expf(pw));
        w0 = al * trans; trans *= (1.f - al);
      }
      {
        float dx = pixx - smx[k+1], dy = pixy - smy[k+1];
        float pw = -0.5f*(sia[k+1]*dx*dx + 2.f*sib[k+1]*dx*dy + sic[k+1]*dy*dy);
        float al = fminf(0.99f, sop[k+1] * __expf(pw));
        w1 = al * trans; trans *= (1.f - al);
      }
      {
        float dx = pixx - smx[k+2], dy = pixy - smy[k+2];
        float pw = -0.5f*(sia[k+2]*dx*dx + 2.f*sib[k+2]*dx*dy + sic[k+2]*dy*dy);
        float al = fminf(0.99f, sop[k+2] * __expf(pw));
        w2 = al * trans; trans *= (1.f - al);
      }
      {
        float dx = pixx - smx[k+3], dy = pixy - smy[k+3];
        float pw = -0.5f*(sia[k+3]*dx*dx + 2.f*sib[k+3]*dx*dy + sic[k+3]*dy*dy);
        float al = fminf(0.99f, sop[k+3] * __expf(pw));
        w3 = al * trans; trans *= (1.f - al);
      }
      int k0 = k + 2*half;
      v2f a, b;
      a.x = half ? w2 : w0;
      a.y = half ? w3 : w1;
      b.x = bsrc[k0];
      b.y = bsrc[k0 + 1];
      acc = __builtin_amdgcn_wmma_f32_16x16x4_f32(
          false, a, false, b, (short)0, acc, false, false);
    }

    // D layout: VGPR r -> M = r (lanes 0-15) / r+8 (lanes 16-31); N = nn
    union { v8f v; float f[8]; } u;
    u.v = acc;
    if (nn < 3) {
#pragma unroll
      for (int r = 0; r < 8; r++) {
        int pxc = r + 8*half;
        int X = (t & (NTX-1))*16 + pxc;
        int Y = (t >> 5)*16 + pb;
        out[(Y*WIDTH + X)*3 + nn] = u.f[r];
      }
    }
  }
}

// ---------------------------------------------------------------------------
extern "C" void kernel_launch(void* const* d_in, const int* in_sizes, int n_in,
                              void* d_out, int out_size, void* d_ws, size_t ws_size,
                              hipStream_t stream) {
  const float* pos   = (const float*)d_in[0];
  const float* rgb   = (const float*)d_in[1];
  const float* opac  = (const float*)d_in[2];
  const float* quat  = (const float*)d_in[3];
  const float* scl   = (const float*)d_in[4];
  const float* w2cq  = (const float*)d_in[5];
  const float* w2ct  = (const float*)d_in[6];
  float* out = (float*)d_out;

  int N = in_sizes[2];  // opacity is (N,)

  char* ws = (char*)d_ws;
  float4* gsel  = (float4*)ws;  ws += (size_t)N * sizeof(float4);
  float4* gA    = (float4*)ws;  ws += (size_t)N * sizeof(float4);
  float*  gIC   = (float*)ws;   ws += (size_t)N * sizeof(float);
  int* sel_idx  = (int*)ws;     ws += (size_t)NTILES * KPT * sizeof(int);
  int* sel_cnt  = (int*)ws;     ws += (size_t)NTILES * sizeof(int);

  k_preprocess<<<(N + 255)/256, 256, 0, stream>>>(pos, quat, scl, w2cq, w2ct, N,
                                                  gsel, gA, gIC);
  k_select<<<NTILES, 256, 0, stream>>>(gsel, N, sel_idx, sel_cnt);
  k_raster<<<NTILES, 256, 0, stream>>>(gA, gIC, rgb, opac, sel_idx, sel_cnt, out);
}